// Message_6648609374628
// MI455X (gfx1250) — compile-verified
//
#include <hip/hip_runtime.h>
#include <math.h>

#define N_ATOMS 50000
#define N_PAIRS 1000000
#define FDIM 64
#define F3 192

typedef float v2f __attribute__((ext_vector_type(2)));
typedef float v8f __attribute__((ext_vector_type(8)));

__device__ __forceinline__ float silu_f(float x) {
    return x / (1.0f + __expf(-x));
}

// ---------------------------------------------------------------------------
// Kernel 1: per-atom MLP  t = silu(s @ W1 + b1) @ W2 + b2
// One wave32 handles 16 atoms (M=16). fp32 WMMA 16x16x4, K=64.
//   A(m,k): lane = m + 16*(k/2), vgpr = k&1
//   B(k,n): lane = n + 16*(k/2), vgpr = k&1
//   C/D(m,n): lane = n + 16*(m/8), vgpr = m&7
// Layer-1 result re-fragmented via per-wave LDS tile (no block barrier needed).
// ---------------------------------------------------------------------------
__global__ void __launch_bounds__(128) mlp_wmma_kernel(
    const float* __restrict__ s,     // [N, 64]
    const float* __restrict__ W1,    // [64, 64]
    const float* __restrict__ b1,    // [64]
    const float* __restrict__ W2,    // [64, 192]
    const float* __restrict__ b2,    // [192]
    float* __restrict__ t)           // [N, 192] (workspace)
{
    __shared__ float h_lds[4 * 16 * FDIM];   // 16KB: 4 waves x (16x64) tile

    const int lane = threadIdx.x & 31;
    const int wave = threadIdx.x >> 5;
    const int tile = blockIdx.x * 4 + wave;
    const int ntiles = N_ATOMS / 16;         // 3125, exact
    if (tile >= ntiles) return;

    const int base = tile * 16;
    const int lm = lane & 15;                // m (A) / n (B,C) selector
    const int lh = lane >> 4;                // K-half / M-half selector
    float* hw = &h_lds[wave * 16 * FDIM];

    // --- load all layer-1 A fragments (16 k-blocks of 4) -------------------
    v2f a1[16];
    const float* srow = s + (size_t)(base + lm) * FDIM;
    #pragma unroll
    for (int kk = 0; kk < 16; ++kk) {
        const float* ap = srow + kk * 4 + 2 * lh;
        v2f a; a.x = ap[0]; a.y = ap[1];
        a1[kk] = a;
    }

    // --- layer 1: 4 output column-tiles of 16 ------------------------------
    for (int nt = 0; nt < 4; ++nt) {
        const int n0 = nt * 16;
        v8f c = {0.f,0.f,0.f,0.f,0.f,0.f,0.f,0.f};
        #pragma unroll
        for (int kk = 0; kk < 16; ++kk) {
            const int k0 = kk * 4 + 2 * lh;
            v2f b;
            b.x = W1[(k0 + 0) * FDIM + n0 + lm];
            b.y = W1[(k0 + 1) * FDIM + n0 + lm];
            c = __builtin_amdgcn_wmma_f32_16x16x4_f32(
                    false, a1[kk], false, b, (short)0, c, false, false);
        }
        const float bias = b1[n0 + lm];
        #pragma unroll
        for (int r = 0; r < 8; ++r) {
            const int m = r + 8 * lh;
            hw[m * FDIM + n0 + lm] = silu_f(c[r] + bias);
        }
    }

    // same-wave LDS producer->consumer: DS ops are in-order, just wait.
    asm volatile("s_wait_dscnt 0" ::: "memory");

    // --- re-fragment h as layer-2 A fragments ------------------------------
    v2f a2[16];
    #pragma unroll
    for (int kk = 0; kk < 16; ++kk) {
        const float* ap = &hw[lm * FDIM + kk * 4 + 2 * lh];
        v2f a; a.x = ap[0]; a.y = ap[1];
        a2[kk] = a;
    }

    // --- layer 2: 12 output column-tiles of 16 -----------------------------
    float* trow = t + (size_t)base * F3;
    for (int nt = 0; nt < 12; ++nt) {
        const int n0 = nt * 16;
        v8f c = {0.f,0.f,0.f,0.f,0.f,0.f,0.f,0.f};
        #pragma unroll
        for (int kk = 0; kk < 16; ++kk) {
            const int k0 = kk * 4 + 2 * lh;
            v2f b;
            b.x = W2[(k0 + 0) * F3 + n0 + lm];
            b.y = W2[(k0 + 1) * F3 + n0 + lm];
            c = __builtin_amdgcn_wmma_f32_16x16x4_f32(
                    false, a2[kk], false, b, (short)0, c, false, false);
        }
        const float bias = b2[n0 + lm];
        #pragma unroll
        for (int r = 0; r < 8; ++r) {
            const int m = r + 8 * lh;
            trow[m * F3 + n0 + lm] = c[r] + bias;
        }
    }
}

// ---------------------------------------------------------------------------
// Kernel 2: pair scatter. One wave32 per pair; lane owns 2 contiguous
// features -> all loads are coalesced b64, scatter via global_atomic_add_f32.
// Streamed-once inputs (W_ij, dir_ij, pairlist) use non-temporal loads so the
// 192MB L2 stays reserved for the reused arrays (t, v gathers + q/mu atomics).
//   ds = W_ij * t[j]  (3F split)
//   q[i]  += ds1
//   mu[i] += ds2*dir + ds3*v[j]
// ---------------------------------------------------------------------------
__global__ void __launch_bounds__(256) pair_kernel(
    const float* __restrict__ W_ij,      // [P, 192]  (streamed once, NT)
    const float* __restrict__ dir_ij,    // [P, 3]    (streamed once, NT)
    const long long* __restrict__ pl,    // [2, P]    (streamed once, NT)
    const float* __restrict__ t,         // [N, 192]  (L2-resident gather)
    const float* __restrict__ v,         // [N, 3, 64](L2-resident gather)
    float* __restrict__ q,               // [N, 64]   (atomic accumulate)
    float* __restrict__ mu)              // [N, 192]  (atomic accumulate)
{
    int p = (int)((blockIdx.x * (unsigned)blockDim.x + threadIdx.x) >> 5);
    if (p >= N_PAIRS) return;
    p = __builtin_amdgcn_readfirstlane(p);   // pair index is wave-uniform
    const int lane = threadIdx.x & 31;
    const int f0 = lane * 2;

    const int i = (int)__builtin_nontemporal_load(pl + p);
    const int j = (int)__builtin_nontemporal_load(pl + N_PAIRS + p);

    const float* wp = W_ij + (size_t)p * F3;
    const float* tj = t + (size_t)j * F3;
    const float* vj = v + (size_t)j * F3;

    const v2f w1 = __builtin_nontemporal_load((const v2f*)(wp +       f0));
    const v2f w2 = __builtin_nontemporal_load((const v2f*)(wp +  64 + f0));
    const v2f w3 = __builtin_nontemporal_load((const v2f*)(wp + 128 + f0));
    const v2f t1 = *(const v2f*)(tj +       f0);
    const v2f t2 = *(const v2f*)(tj +  64 + f0);
    const v2f t3 = *(const v2f*)(tj + 128 + f0);

    const float d0 = __builtin_nontemporal_load(dir_ij + (size_t)p * 3 + 0);
    const float d1 = __builtin_nontemporal_load(dir_ij + (size_t)p * 3 + 1);
    const float d2 = __builtin_nontemporal_load(dir_ij + (size_t)p * 3 + 2);

    const float ds1x = w1.x * t1.x, ds1y = w1.y * t1.y;
    const float ds2x = w2.x * t2.x, ds2y = w2.y * t2.y;
    const float ds3x = w3.x * t3.x, ds3y = w3.y * t3.y;

    float* qi = q + (size_t)i * FDIM + f0;
    atomicAdd(qi + 0, ds1x);
    atomicAdd(qi + 1, ds1y);

    const float dir3[3] = {d0, d1, d2};
    float* mui = mu + (size_t)i * F3 + f0;
    #pragma unroll
    for (int d = 0; d < 3; ++d) {
        const v2f vv = *(const v2f*)(vj + d * 64 + f0);
        const float mx = ds2x * dir3[d] + ds3x * vv.x;
        const float my = ds2y * dir3[d] + ds3y * vv.y;
        atomicAdd(mui + d * 64 + 0, mx);
        atomicAdd(mui + d * 64 + 1, my);
    }
}

extern "C" void kernel_launch(void* const* d_in, const int* in_sizes, int n_in,
                              void* d_out, int out_size, void* d_ws, size_t ws_size,
                              hipStream_t stream) {
    const float*     s   = (const float*)d_in[0];      // [N,1,64]
    const float*     v   = (const float*)d_in[1];      // [N,3,64]
    const float*     Wij = (const float*)d_in[2];      // [P,192]
    const float*     dir = (const float*)d_in[3];      // [P,3]
    const long long* pl  = (const long long*)d_in[4];  // [2,P] int64
    const float*     W1  = (const float*)d_in[5];      // [64,64]
    const float*     b1  = (const float*)d_in[6];      // [64]
    const float*     W2  = (const float*)d_in[7];      // [64,192]
    const float*     b2  = (const float*)d_in[8];      // [192]

    float* q  = (float*)d_out;                              // [N,64]
    float* mu = (float*)d_out + (size_t)N_ATOMS * FDIM;     // [N,192]
    float* t  = (float*)d_ws;                               // [N,192] = 38.4 MB

    // Re-initialize outputs every call (atomics accumulate on top).
    hipMemcpyAsync(q,  s, (size_t)N_ATOMS * FDIM * sizeof(float),
                   hipMemcpyDeviceToDevice, stream);
    hipMemcpyAsync(mu, v, (size_t)N_ATOMS * F3 * sizeof(float),
                   hipMemcpyDeviceToDevice, stream);

    const int ntiles  = N_ATOMS / 16;            // 3125
    const int blocks1 = (ntiles + 3) / 4;        // 4 waves/block
    mlp_wmma_kernel<<<blocks1, 128, 0, stream>>>(s, W1, b1, W2, b2, t);

    const int wpb     = 256 / 32;                // waves per block
    const int blocks2 = (N_PAIRS + wpb - 1) / wpb;
    pair_kernel<<<blocks2, 256, 0, stream>>>(Wij, dir, pl, t, v, q, mu);
}